// SigLoss_72825465471553
// MI455X (gfx1250) — compile-verified
//
#include <hip/hip_runtime.h>

typedef float v2f __attribute__((ext_vector_type(2)));
typedef float v8f __attribute__((ext_vector_type(8)));

#define NPATH 32
#define LEN   128
#define DIM   8
#define Q     127          // increments per path (rows/cols of inc)
#define TILES 8            // 128/16 tiles along each dimension
#define NBLOCKS (3 * NPATH * NPATH)   // XX, YY, XY pair blocks

// One wave (32 threads) per path pair. Computes inc = dA * dB^T tile-by-tile
// with V_WMMA_F32_16X16X4_F32 (fp32, K=8 as two chained K=4 steps) into an
// LDS strip, then runs the Goursat recurrence rows with a wave prefix scan.
__global__ __launch_bounds__(32) void sig_pair_kernel(const float* __restrict__ X,
                                                      const float* __restrict__ Y,
                                                      float* __restrict__ out) {
    __shared__ __align__(16) float sA[LEN * DIM];   // dA, row 127 zero-padded
    __shared__ __align__(16) float sB[LEN * DIM];   // dB, row 127 zero-padded
    __shared__ __align__(16) float sInc[16 * 128];  // one 16-row strip of inc

    const int lane = threadIdx.x;                   // 0..31 (wave32)
    const int p    = blockIdx.x;
    const int blk  = p >> 10;                       // 0:XX 1:YY 2:XY
    const int idx  = p & 1023;
    const int a0   = idx >> 5;
    const int b0   = idx & 31;

    const float* pA = (blk == 1) ? (Y + a0 * LEN * DIM) : (X + a0 * LEN * DIM);
    const float* pB = (blk == 0) ? (X + b0 * LEN * DIM) : (Y + b0 * LEN * DIM);
    const float  wgt = ((blk == 2) ? -2.0f : 1.0f) * (1.0f / 1024.0f);

    // ---- Build increment matrices dA, dB in LDS (row 127 = 0 padding) ----
    for (int r = lane; r < LEN; r += 32) {
        float4* sa = reinterpret_cast<float4*>(&sA[r * DIM]);
        float4* sb = reinterpret_cast<float4*>(&sB[r * DIM]);
        if (r < Q) {
            const float4* qa0 = reinterpret_cast<const float4*>(pA + r * DIM);
            const float4* qa1 = reinterpret_cast<const float4*>(pA + (r + 1) * DIM);
            const float4* qb0 = reinterpret_cast<const float4*>(pB + r * DIM);
            const float4* qb1 = reinterpret_cast<const float4*>(pB + (r + 1) * DIM);
            float4 u0 = qa0[0], u1 = qa0[1], v0 = qa1[0], v1 = qa1[1];
            float4 w0 = qb0[0], w1 = qb0[1], x0 = qb1[0], x1 = qb1[1];
            sa[0] = make_float4(v0.x - u0.x, v0.y - u0.y, v0.z - u0.z, v0.w - u0.w);
            sa[1] = make_float4(v1.x - u1.x, v1.y - u1.y, v1.z - u1.z, v1.w - u1.w);
            sb[0] = make_float4(x0.x - w0.x, x0.y - w0.y, x0.z - w0.z, x0.w - w0.w);
            sb[1] = make_float4(x1.x - w1.x, x1.y - w1.y, x1.z - w1.z, x1.w - w1.w);
        } else {
            float4 z = make_float4(0.f, 0.f, 0.f, 0.f);
            sa[0] = z; sa[1] = z; sb[0] = z; sb[1] = z;
        }
    }
    __syncthreads();

    // ---- Preload all B fragments (column tiles; reused by every row strip) ----
    // A/B operand layout for V_WMMA_F32_16X16X4_F32 (wave32):
    //   lanes 0-15: K = {0,1}; lanes 16-31: K = {2,3}; lane%16 = M (A) / N (B).
    const int n  = lane & 15;
    const int kb = (lane >> 4) * 2;
    v2f bf[TILES][2];
#pragma unroll
    for (int t = 0; t < TILES; ++t) {
        const float* row = &sB[(t * 16 + n) * DIM];
        bf[t][0] = (v2f){row[kb],     row[kb + 1]};
        bf[t][1] = (v2f){row[kb + 4], row[kb + 5]};
    }

    // ---- Recurrence state: lane l holds prev K-row entries [4l .. 4l+3] ----
    float p0 = 1.0f, p1 = 1.0f, p2 = 1.0f, p3 = 1.0f;

    for (int s = 0; s < TILES; ++s) {
        // A fragment for this 16-row strip
        const float* arow = &sA[(s * 16 + n) * DIM];
        v2f afr0 = (v2f){arow[kb],     arow[kb + 1]};
        v2f afr1 = (v2f){arow[kb + 4], arow[kb + 5]};

        // inc strip = dA[16s:16s+16, :] * dB^T via WMMA, scattered to LDS
        const int rbase = (lane >> 4) * 8;   // C VGPR r -> local row r + rbase
#pragma unroll
        for (int t = 0; t < TILES; ++t) {
            v8f c = {};
            c = __builtin_amdgcn_wmma_f32_16x16x4_f32(
                    false, afr0, false, bf[t][0], (short)0, c, false, false);
            c = __builtin_amdgcn_wmma_f32_16x16x4_f32(
                    false, afr1, false, bf[t][1], (short)0, c, false, false);
            const int col = t * 16 + n;
#pragma unroll
            for (int r = 0; r < 8; ++r)
                sInc[(rbase + r) * 128 + col] = c[r];
        }
        __syncthreads();

        // Goursat rows of this strip (last strip has 15 valid rows: i <= 126)
        const int rows = (s == TILES - 1) ? 15 : 16;
        for (int il = 0; il < rows; ++il) {
            float4 q4 = *reinterpret_cast<const float4*>(&sInc[il * 128 + lane * 4]);
            float pn = __shfl_down(p0, 1);          // prev[4*lane + 4]
            float d0 = p1 + p0 * (q4.x - 1.0f);
            float d1 = p2 + p1 * (q4.y - 1.0f);
            float d2 = p3 + p2 * (q4.z - 1.0f);
            float d3 = (lane == 31) ? 0.0f : (pn + p3 * (q4.w - 1.0f));
            // lane-local exclusive prefix + lane total
            float e1 = d0, e2 = e1 + d1, e3 = e2 + d2;
            float T  = e3 + d3;
            // wave32 inclusive scan of lane totals
            float v = T;
#pragma unroll
            for (int off = 1; off < 32; off <<= 1) {
                float tsh = __shfl_up(v, off);
                if (lane >= off) v += tsh;
            }
            float E    = v - T;                     // exclusive lane prefix
            float base = 1.0f + E;                  // next[4l+t] = 1 + E + e_t
            p0 = base; p1 = base + e1; p2 = base + e2; p3 = base + e3;
        }
        __syncthreads();
    }

    float res = __shfl(p3, 31);                     // K[127][127]
    if (lane == 0) out[p] = wgt * res;
}

// Deterministic tree reduction of the 3072 weighted pair results -> mean loss.
__global__ __launch_bounds__(128) void sig_reduce_kernel(const float* __restrict__ ws,
                                                         float* __restrict__ out) {
    __shared__ float red[128];
    const int tid = threadIdx.x;
    float s = 0.0f;
    for (int i = tid; i < NBLOCKS; i += 128) s += ws[i];
    red[tid] = s;
    __syncthreads();
    for (int w = 64; w > 0; w >>= 1) {
        if (tid < w) red[tid] += red[tid + w];
        __syncthreads();
    }
    if (tid == 0) out[0] = red[0];
}

extern "C" void kernel_launch(void* const* d_in, const int* in_sizes, int n_in,
                              void* d_out, int out_size, void* d_ws, size_t ws_size,
                              hipStream_t stream) {
    const float* X = (const float*)d_in[0];
    const float* Y = (const float*)d_in[1];
    float* ws = (float*)d_ws;                       // 3072 floats of scratch

    sig_pair_kernel<<<NBLOCKS, 32, 0, stream>>>(X, Y, ws);
    sig_reduce_kernel<<<1, 128, 0, stream>>>(ws, (float*)d_out);
}